// EPN_layer_39195871543872
// MI455X (gfx1250) — compile-verified
//
#include <hip/hip_runtime.h>

// EPN pairwise-MLP kernel for gfx1250 (MI455X), fp32 WMMA 16x16x4 path.
//
// Factorization: layer1(z@W1) = A32[i] + B32[j] + e@W1e (+b1), so the O(N^2)
// loop only pays the e-projection (K=8) plus layers 2/3. b3 cancels in the
// antisymmetric difference.

typedef __attribute__((ext_vector_type(2))) float v2f;
typedef __attribute__((ext_vector_type(8))) float v8f;

#define EPN_TOL 1e-5f

__device__ __forceinline__ v8f splat8(float v) {
  v8f r;
#pragma unroll
  for (int k = 0; k < 8; ++k) r[k] = v;
  return r;
}

__device__ __forceinline__ v8f wmma4(v2f a, v2f b, v8f c) {
#if __has_builtin(__builtin_amdgcn_wmma_f32_16x16x4_f32)
  // 8 args: (neg_a, A, neg_b, B, c_mod, C, reuse_a, reuse_b)
  return __builtin_amdgcn_wmma_f32_16x16x4_f32(false, a, false, b, (short)0, c,
                                               false, false);
#else
  v8f r;
#pragma unroll
  for (int k = 0; k < 8; ++k) r[k] = c[k] + a[0] * b[0] + a[1] * b[1];
  return r;
#endif
}

// B=8, N=256, DH=32, DX=3, DQ=1, DE=8, F=36, H1=H2=32, DIN=80
__global__ __launch_bounds__(256) void epn_kernel(
    const float* __restrict__ hbuf, const float* __restrict__ ebuf,
    const float* __restrict__ xbuf, const float* __restrict__ qbuf,
    const float* __restrict__ mbuf, const float* __restrict__ W1,
    const float* __restrict__ b1, const float* __restrict__ W2,
    const float* __restrict__ b2, const float* __restrict__ W3,
    float* __restrict__ out) {
  extern __shared__ float smem[];
  float* sA  = smem;                 // [256][33]  A32 = inp_atom @ W1[0:36]
  float* sB  = smem + 256 * 33;      // [256][33]  B32 = inp_atom @ W1[36:72]
  float* uT  = smem + 2 * 256 * 33;  // [8 waves][16][34] activation bounce
  float* sdf = uT + 8 * 16 * 34;     // [8][16] per-row layer3 dots
  float* acc = sdf + 8 * 16;         // [8][16] per-wave, per-i accumulators

  const int tid  = threadIdx.x;
  const int w    = tid >> 5;   // wave id (8 waves)
  const int lane = tid & 31;   // wave32
  const int hh   = lane >> 4;  // half-wave
  const int ll   = lane & 15;
  const int bidx = blockIdx.x >> 4;  // batch
  const int it   = blockIdx.x & 15;  // i tile
  const int ibase = it * 16;

  // ---- per-atom first-layer projections (all 256 atoms of this batch) ----
  {
    const int n = tid;  // one atom per thread
    float feat[36];
    const float* xp = xbuf + (bidx * 256 + n) * 3;
    const float* hp = hbuf + (bidx * 256 + n) * 32;
    feat[0] = xp[0];
    feat[1] = xp[1];
    feat[2] = xp[2];
#pragma unroll
    for (int f = 0; f < 32; ++f) feat[3 + f] = hp[f];
    feat[35] = qbuf[bidx * 256 + n];
#pragma unroll 4
    for (int c = 0; c < 32; ++c) {
      float a = 0.f, b = 0.f;
#pragma unroll
      for (int f = 0; f < 36; ++f) {
        a += feat[f] * W1[f * 32 + c];
        b += feat[f] * W1[(36 + f) * 32 + c];
      }
      sA[n * 33 + c] = a;
      sB[n * 33 + c] = b;
    }
  }
  if (tid < 128) acc[tid] = 0.f;
  __syncthreads();

  // ---- loop-invariant WMMA B-tiles in registers ----
  // B(4x16) layout assumption: reg r, lane L -> (K = r + 2*(L/16), N = L%16)
  v2f bE[2][2];    // W1 e-rows 72..79: 2 K-steps x 2 N-tiles
  v2f bW2t[8][2];  // W2 (32x32): 8 K-steps x 2 N-tiles
#pragma unroll
  for (int s = 0; s < 2; ++s)
#pragma unroll
    for (int nt = 0; nt < 2; ++nt) {
      bE[s][nt][0] = W1[(72 + s * 4 + 2 * hh + 0) * 32 + nt * 16 + ll];
      bE[s][nt][1] = W1[(72 + s * 4 + 2 * hh + 1) * 32 + nt * 16 + ll];
    }
#pragma unroll
  for (int s = 0; s < 8; ++s)
#pragma unroll
    for (int nt = 0; nt < 2; ++nt) {
      bW2t[s][nt][0] = W2[(s * 4 + 2 * hh + 0) * 32 + nt * 16 + ll];
      bW2t[s][nt][1] = W2[(s * 4 + 2 * hh + 1) * 32 + nt * 16 + ll];
    }
  const float b1c[2] = {b1[ll], b1[16 + ll]};
  const float b2c[2] = {b2[ll], b2[16 + ll]};
  const float w3c[2] = {W3[ll], W3[16 + ll]};

  float* uW = uT + w * (16 * 34);  // wave-private bounce tile, stride 34

  for (int jt = w; jt < 16; jt += 8) {
    const int jbase = jt * 16;
    // Per-row (D-layout: row = r + 8*hh) A32[j]/B32[j] for this j tile.
    float AjD[2][8], BjD[2][8];
#pragma unroll
    for (int r = 0; r < 8; ++r) {
      const int jr = jbase + r + 8 * hh;
#pragma unroll
      for (int nt = 0; nt < 2; ++nt) {
        AjD[nt][r] = sA[jr * 33 + nt * 16 + ll];
        BjD[nt][r] = sB[jr * 33 + nt * 16 + ll];
      }
    }
#pragma unroll 1
    for (int il = 0; il < 16; ++il) {
      const int i = ibase + il;
      // e as WMMA-A (16 j-rows x 8 K). A layout: (M = L%16, K = s*4+2*hh+r).
      const float* ep = ebuf + (((size_t)bidx * 256 + i) * 256 + jbase) * 8;
      const v2f a0 = *(const v2f*)(ep + ll * 8 + 2 * hh);      // K-step 0
      const v2f a1 = *(const v2f*)(ep + ll * 8 + 4 + 2 * hh);  // K-step 1

      // Gate: is_near = (max_k e > TOL); row j's 8 values live in lanes
      // j and j+16, so fold with an xor-16 shuffle. m = mask[b,i,j,0].
      float lm = fmaxf(fmaxf(a0[0], a0[1]), fmaxf(a1[0], a1[1]));
      const float om = fmaxf(lm, __shfl_xor(lm, 16, 32));
      const float nearv = om > EPN_TOL ? 1.f : 0.f;
      const float mv = mbuf[((size_t)bidx * 256 + i) * 256 + jbase + ll];
      const float g = 0.5f * mv * nearv;

      // Layer1 e-term: CE = e @ W1e + b1 (16 j-rows x 32 cols).
      v8f CE[2] = {splat8(b1c[0]), splat8(b1c[1])};
      CE[0] = wmma4(a0, bE[0][0], CE[0]);
      CE[0] = wmma4(a1, bE[1][0], CE[0]);
      CE[1] = wmma4(a0, bE[0][1], CE[1]);
      CE[1] = wmma4(a1, bE[1][1], CE[1]);

      const float aib[2] = {sA[i * 33 + ll], sA[i * 33 + 16 + ll]};
      const float bib[2] = {sB[i * 33 + ll], sB[i * 33 + 16 + ll]};

      // ---- direction ij: u = relu(CE + A[i] + B[j]) ----
#pragma unroll
      for (int nt = 0; nt < 2; ++nt)
#pragma unroll
        for (int r = 0; r < 8; ++r) {
          const float u = CE[nt][r] + aib[nt] + BjD[nt][r];
          uW[(r + 8 * hh) * 34 + nt * 16 + ll] = fmaxf(u, 0.f);
        }
      v8f C2a[2] = {splat8(b2c[0]), splat8(b2c[1])};
#pragma unroll
      for (int s = 0; s < 8; ++s) {
        const v2f av = *(const v2f*)(uW + ll * 34 + s * 4 + 2 * hh);
        C2a[0] = wmma4(av, bW2t[s][0], C2a[0]);
        C2a[1] = wmma4(av, bW2t[s][1], C2a[1]);
      }

      // ---- direction ji: u = relu(CE + A[j] + B[i]) ----
#pragma unroll
      for (int nt = 0; nt < 2; ++nt)
#pragma unroll
        for (int r = 0; r < 8; ++r) {
          const float u = CE[nt][r] + AjD[nt][r] + bib[nt];
          uW[(r + 8 * hh) * 34 + nt * 16 + ll] = fmaxf(u, 0.f);
        }
      v8f C2b[2] = {splat8(b2c[0]), splat8(b2c[1])};
#pragma unroll
      for (int s = 0; s < 8; ++s) {
        const v2f av = *(const v2f*)(uW + ll * 34 + s * 4 + 2 * hh);
        C2b[0] = wmma4(av, bW2t[s][0], C2b[0]);
        C2b[1] = wmma4(av, bW2t[s][1], C2b[1]);
      }

      // ---- layer3 + antisym: s_ij - s_ji = (relu(vij)-relu(vji)) @ W3 ----
#pragma unroll
      for (int r = 0; r < 8; ++r) {
        const float d0 = fmaxf(C2a[0][r], 0.f) - fmaxf(C2b[0][r], 0.f);
        const float d1 = fmaxf(C2a[1][r], 0.f) - fmaxf(C2b[1][r], 0.f);
        float t = d0 * w3c[0] + d1 * w3c[1];
        t += __shfl_xor(t, 1, 32);
        t += __shfl_xor(t, 2, 32);
        t += __shfl_xor(t, 4, 32);
        t += __shfl_xor(t, 8, 32);  // half-wave sum: rows r and r+8
        if (ll == 0) sdf[w * 16 + r + 8 * hh] = t;
      }
      const float sd = sdf[w * 16 + ll];  // per-lane row dot (j = jbase+ll)
      float val = g * sd;
      val += __shfl_xor(val, 1, 32);
      val += __shfl_xor(val, 2, 32);
      val += __shfl_xor(val, 4, 32);
      val += __shfl_xor(val, 8, 32);  // sum over 16 j's
      if (lane == 0) acc[w * 16 + il] += val;
    }
  }

  __syncthreads();
  if (tid < 16) {
    float s = 0.f;
#pragma unroll
    for (int ww = 0; ww < 8; ++ww) s += acc[ww * 16 + tid];
    out[bidx * 256 + ibase + tid] = qbuf[bidx * 256 + ibase + tid] + s;
  }
}

extern "C" void kernel_launch(void* const* d_in, const int* in_sizes, int n_in,
                              void* d_out, int out_size, void* d_ws,
                              size_t ws_size, hipStream_t stream) {
  (void)in_sizes;
  (void)n_in;
  (void)out_size;
  (void)d_ws;
  (void)ws_size;
  const float* h = (const float*)d_in[0];
  const float* e = (const float*)d_in[1];
  const float* x = (const float*)d_in[2];
  const float* q = (const float*)d_in[3];
  const float* m = (const float*)d_in[4];
  const float* W1 = (const float*)d_in[5];
  const float* b1 = (const float*)d_in[6];
  const float* W2 = (const float*)d_in[7];
  const float* b2 = (const float*)d_in[8];
  const float* W3 = (const float*)d_in[9];
  // b3 (d_in[10]) cancels in the antisymmetric difference.
  const size_t smem_bytes =
      (size_t)(2 * 256 * 33 + 8 * 16 * 34 + 128 + 128) * sizeof(float);
  // ~86 KB dynamic LDS: above the default cap, well within gfx1250's 320 KB.
  // Host-side attribute set; not a stream/alloc op, so graph-capture safe.
  (void)hipFuncSetAttribute((const void*)epn_kernel,
                            hipFuncAttributeMaxDynamicSharedMemorySize,
                            (int)smem_bytes);
  epn_kernel<<<dim3(8 * 16), dim3(256), smem_bytes, stream>>>(
      h, e, x, q, m, W1, b1, W2, b2, W3, (float*)d_out);
}